// GCN1_31507880083906
// MI455X (gfx1250) — compile-verified
//
#include <hip/hip_runtime.h>

typedef float v2f __attribute__((ext_vector_type(2)));
typedef float v8f __attribute__((ext_vector_type(8)));

// ---------------------------------------------------------------------------
// K0: init accumulators. deg starts at 1.0 (self loop), layer-2 accum at 0.
// ---------------------------------------------------------------------------
__global__ void k_init(float* __restrict__ dinv, float* __restrict__ gagg, int n) {
  int i = blockIdx.x * blockDim.x + threadIdx.x;
  if (i < n) {
    dinv[i]      = 1.0f;   // self-loop contribution to degree
    gagg[2*i]    = 0.0f;
    gagg[2*i+1]  = 0.0f;
  }
}

// ---------------------------------------------------------------------------
// K1: degree count. 4 edges/thread via int4 (global_load_b128) index loads.
// ---------------------------------------------------------------------------
__global__ void k_degree(const int* __restrict__ dstIdx, float* __restrict__ deg, int e) {
  int i = (blockIdx.x * blockDim.x + threadIdx.x) * 4;
  if (i + 3 < e) {
    int4 d = *(const int4*)(dstIdx + i);
    atomicAdd(&deg[d.x], 1.0f);
    atomicAdd(&deg[d.y], 1.0f);
    atomicAdd(&deg[d.z], 1.0f);
    atomicAdd(&deg[d.w], 1.0f);
  } else {
    for (int j = i; j < e; ++j) atomicAdd(&deg[dstIdx[j]], 1.0f);
  }
}

// ---------------------------------------------------------------------------
// K2: dinv = deg^-1/2 in place; seed s with the self-loop term dinv^2 * x[i]
// ---------------------------------------------------------------------------
__global__ void k_dinv(const float* __restrict__ x, float* __restrict__ dinv,
                       float* __restrict__ s, int n) {
  int i = blockIdx.x * blockDim.x + threadIdx.x;
  if (i < n) {
    float dv = rsqrtf(dinv[i]);   // deg >= 1 always (self loop)
    dinv[i] = dv;
    s[i] = dv * dv * x[i];
  }
}

// ---------------------------------------------------------------------------
// K3: layer-1 edge scatter (scalar feature, factored out of the 32-wide
// aggregation since lin precedes propagate and in-dim == 1). 4 edges/thread.
// ---------------------------------------------------------------------------
__global__ void k_scatter1(const int* __restrict__ srcIdx, const int* __restrict__ dstIdx,
                           const float* __restrict__ x, const float* __restrict__ dinv,
                           float* __restrict__ s, int e) {
  int i = (blockIdx.x * blockDim.x + threadIdx.x) * 4;
  if (i + 3 < e) {
    int4 sv = *(const int4*)(srcIdx + i);
    int4 dv = *(const int4*)(dstIdx + i);
    atomicAdd(&s[dv.x], dinv[sv.x] * dinv[dv.x] * x[sv.x]);
    atomicAdd(&s[dv.y], dinv[sv.y] * dinv[dv.y] * x[sv.y]);
    atomicAdd(&s[dv.z], dinv[sv.z] * dinv[dv.z] * x[sv.z]);
    atomicAdd(&s[dv.w], dinv[sv.w] * dinv[dv.w] * x[sv.w]);
  } else {
    for (int j = i; j < e; ++j) {
      int sn = srcIdx[j], dn = dstIdx[j];
      atomicAdd(&s[dn], dinv[sn] * dinv[dn] * x[sn]);
    }
  }
}

// ---------------------------------------------------------------------------
// K4: fused node transform via WMMA f32 16x16x4 (full fp32 precision):
//   h1[n][0:32] = relu(s[n]*W1 + b1)          (A tile, 16 nodes x 32 feats)
//   g[n][0:2]   = h1[n] @ W2                  (B = W2 zero-padded to 32x16)
// One wave per 16-node tile; K=32 accumulated as 8 WMMA steps of K=4.
// B fragments are built branch-free (clamped index + lane mask) and hoisted
// out of the WMMA loop so no EXEC save/restore appears between WMMAs.
// ---------------------------------------------------------------------------
__global__ __launch_bounds__(256)
void k_wmma_transform(const float* __restrict__ s,
                      const float* __restrict__ W1,   // 32 floats ([1,32])
                      const float* __restrict__ b1,   // 32 floats
                      const float* __restrict__ W2,   // [32,2] row-major
                      float* __restrict__ g, int n, int ntiles) {
  int lane = threadIdx.x & 31;
  int wave = blockIdx.x * (blockDim.x >> 5) + (threadIdx.x >> 5);
  if (wave >= ntiles) return;            // wave-uniform: EXEC stays all-1s
  int t0   = wave * 16;
  int half = lane >> 4;                  // 0: K%4 in {0,1}; 1: K%4 in {2,3}
  int m    = lane & 15;
  int node = t0 + m;
  // Clamped load + zero-select instead of divergent load (branch-free).
  int nidx   = node < n ? node : 0;
  float sn   = (node < n) ? s[nidx] : 0.0f;
  float cmsk = (m < 2) ? 1.0f : 0.0f;    // B columns >= 2 are zero padding
  int   cm   = (m < 2) ? m : 0;          // always-in-bounds W2 column

  // Preload all B fragments (W2 padded to 32x16) — 16 unconditional loads,
  // hoisted above the WMMA chain by the compiler.
  v2f Bf[8];
  #pragma unroll
  for (int i = 0; i < 8; ++i) {
    int f0 = 4 * i + 2 * half;
    Bf[i].x = W2[(f0    ) * 2 + cm] * cmsk;
    Bf[i].y = W2[(f0 + 1) * 2 + cm] * cmsk;
  }

  v8f c = {};
  #pragma unroll
  for (int i = 0; i < 8; ++i) {
    int f0 = 4 * i + 2 * half;           // feature index for A/B vgpr0
    int f1 = f0 + 1;                     // feature index for A/B vgpr1
    // A fragment: h1 values for (node, f0/f1)
    v2f A;
    A.x = fmaxf(fmaf(sn, W1[f0], b1[f0]), 0.0f);
    A.y = fmaxf(fmaf(sn, W1[f1], b1[f1]), 0.0f);
    c = __builtin_amdgcn_wmma_f32_16x16x4_f32(
            /*neg_a=*/false, A, /*neg_b=*/false, Bf[i],
            /*c_mod=*/(short)0, c, /*reuse_a=*/false, /*reuse_b=*/false);
  }

  // D layout: lane L, vgpr v -> (M = v + 8*(L>=16), Ncol = L&15).
  // Only columns 0,1 are real outputs (lanes 0,1 and 16,17).
  if (m < 2) {
    #pragma unroll
    for (int v = 0; v < 8; ++v) {
      int row = t0 + v + 8 * half;
      if (row < n) g[row * 2 + m] = c[v];
    }
  }
}

// ---------------------------------------------------------------------------
// K5: layer-2 edge scatter of the 2-wide projected features. 2 edges/thread
// with int2 index loads; feature pair loaded as one float2 (8B).
// ---------------------------------------------------------------------------
__global__ void k_scatter2(const int* __restrict__ srcIdx, const int* __restrict__ dstIdx,
                           const float* __restrict__ g, const float* __restrict__ dinv,
                           float* __restrict__ gagg, int e) {
  int i = (blockIdx.x * blockDim.x + threadIdx.x) * 2;
  if (i + 1 < e) {
    int2 sv = *(const int2*)(srcIdx + i);
    int2 dv = *(const int2*)(dstIdx + i);
    float  w0 = dinv[sv.x] * dinv[dv.x];
    float  w1 = dinv[sv.y] * dinv[dv.y];
    float2 g0 = *(const float2*)(g + 2 * sv.x);
    float2 g1 = *(const float2*)(g + 2 * sv.y);
    atomicAdd(&gagg[2*dv.x],     w0 * g0.x);
    atomicAdd(&gagg[2*dv.x + 1], w0 * g0.y);
    atomicAdd(&gagg[2*dv.y],     w1 * g1.x);
    atomicAdd(&gagg[2*dv.y + 1], w1 * g1.y);
  } else if (i < e) {
    int sn = srcIdx[i], dn = dstIdx[i];
    float w = dinv[sn] * dinv[dn];
    float2 gv = *(const float2*)(g + 2 * sn);
    atomicAdd(&gagg[2*dn],     w * gv.x);
    atomicAdd(&gagg[2*dn + 1], w * gv.y);
  }
}

// ---------------------------------------------------------------------------
// K6: add self-loop term + bias, then 2-class log_softmax.
// ---------------------------------------------------------------------------
__global__ void k_finalize(const float* __restrict__ gagg, const float* __restrict__ g,
                           const float* __restrict__ dinv, const float* __restrict__ b2,
                           float* __restrict__ out, int n) {
  int i = blockIdx.x * blockDim.x + threadIdx.x;
  if (i < n) {
    float d2 = dinv[i] * dinv[i];
    float2 ga = *(const float2*)(gagg + 2 * i);
    float2 gv = *(const float2*)(g + 2 * i);
    float o0 = ga.x + d2 * gv.x + b2[0];
    float o1 = ga.y + d2 * gv.y + b2[1];
    float mx  = fmaxf(o0, o1);
    float lse = mx + logf(expf(o0 - mx) + expf(o1 - mx));
    float2 r; r.x = o0 - lse; r.y = o1 - lse;
    *(float2*)(out + 2 * i) = r;
  }
}

// ---------------------------------------------------------------------------
extern "C" void kernel_launch(void* const* d_in, const int* in_sizes, int n_in,
                              void* d_out, int out_size, void* d_ws, size_t ws_size,
                              hipStream_t stream) {
  const float* x  = (const float*)d_in[0];
  const int*   ei = (const int*)d_in[1];   // edge_index [2, E]
  const float* W1 = (const float*)d_in[2];
  const float* b1 = (const float*)d_in[3];
  const float* W2 = (const float*)d_in[4];
  const float* b2 = (const float*)d_in[5];
  float* out = (float*)d_out;

  int n = in_sizes[0];          // x is [N,1]
  int e = in_sizes[1] / 2;
  const int* srcIdx = ei;
  const int* dstIdx = ei + e;

  // Scratch layout: dinv[n] | s[n] | g[2n] | gagg[2n]  (6n floats = 2.4 MB)
  float* dinv = (float*)d_ws;
  float* s    = dinv + n;
  float* g    = s + n;
  float* gagg = g + 2 * (size_t)n;

  const int tb = 256;
  int gn  = (n + tb - 1) / tb;
  int ge4 = ((e + 3) / 4 + tb - 1) / tb;   // 4 edges per thread
  int ge2 = ((e + 1) / 2 + tb - 1) / tb;   // 2 edges per thread

  k_init    <<<gn,  tb, 0, stream>>>(dinv, gagg, n);
  k_degree  <<<ge4, tb, 0, stream>>>(dstIdx, dinv, e);
  k_dinv    <<<gn,  tb, 0, stream>>>(x, dinv, s, n);
  k_scatter1<<<ge4, tb, 0, stream>>>(srcIdx, dstIdx, x, dinv, s, e);

  int ntiles = (n + 15) / 16;
  int gw = (ntiles + 7) / 8;    // 8 waves (tiles) per 256-thread block
  k_wmma_transform<<<gw, tb, 0, stream>>>(s, W1, b1, W2, g, n, ntiles);

  k_scatter2<<<ge2, tb, 0, stream>>>(srcIdx, dstIdx, g, dinv, gagg, e);
  k_finalize<<<gn,  tb, 0, stream>>>(gagg, g, dinv, b2, out, n);
}